// HierNet_42777874268530
// MI455X (gfx1250) — compile-verified
//
#include <hip/hip_runtime.h>
#include <hip/hip_bf16.h>
#include <math.h>

typedef __attribute__((ext_vector_type(2))) float v2f;
typedef __attribute__((ext_vector_type(8))) float v8f;

#define BB   32
#define NN   2048
#define EE   32768
#define CINW 128
#define HF   64
#define EB   (BB * EE)   // 1048576 edges total (fixed-size masked edge list)

static inline int cdiv(long a, int b) { return (int)((a + b - 1) / b); }

// ---------------------------------------------------------------- utilities
__global__ void k_fill(float* __restrict__ p, float v, long n) {
    long i = (long)blockIdx.x * blockDim.x + threadIdx.x;
    if (i < n) p[i] = v;
}

__global__ void k_edges(const int* __restrict__ ei, int* __restrict__ src,
                        int* __restrict__ dst, float* __restrict__ m) {
    long i = (long)blockIdx.x * blockDim.x + threadIdx.x;
    if (i >= EB) return;
    int b = (int)(i / EE);
    src[i] = ei[i] - b * NN;
    dst[i] = ei[(long)EB + i] - b * NN;
    m[i] = 1.0f;
}

// ------------------------------------------------- fp32 WMMA GEMM: C = A@W
// A: [M,K] row-major, W: [K,64] row-major, C: [M,64]. M % 32 == 0, K % 4 == 0.
// One wave -> one 16x16 tile, 8 waves/block -> 32 rows x 64 cols per block.
__global__ __launch_bounds__(256) void k_gemm_wmma(const float* __restrict__ A,
                                                   const float* __restrict__ W,
                                                   float* __restrict__ C,
                                                   int M, int K) {
    int wave  = threadIdx.x >> 5;
    int lane  = threadIdx.x & 31;
    int l15   = lane & 15;
    int hi    = lane >> 4;                 // 0: lanes 0-15, 1: lanes 16-31
    int mbase = blockIdx.x * 32 + (wave >> 2) * 16;
    int nbase = (wave & 3) * 16;
    if (mbase >= M) return;

    const float* arow = A + (long)(mbase + l15) * K;
    v8f acc = {};
    for (int k = 0; k < K; k += 4) {
        int ka = k + hi * 2;
        // A fragment: lane l<16 holds A[m=l, ka..ka+1]; lane>=16 holds K+2..K+3
        v2f a = *(const v2f*)(arow + ka);
        // B fragment (4x16): same K split, N = nbase + l15
        v2f b;
        b.x = W[ka * HF + nbase + l15];
        b.y = W[(ka + 1) * HF + nbase + l15];
        acc = __builtin_amdgcn_wmma_f32_16x16x4_f32(
            /*neg_a=*/false, a, /*neg_b=*/false, b,
            /*c_mod=*/(short)0, acc, /*reuse_a=*/false, /*reuse_b=*/false);
    }
    // C/D layout: VGPR v -> row mbase + v + hi*8, col nbase + l15
    int col = nbase + l15;
    int r0  = mbase + hi * 8;
#pragma unroll
    for (int v = 0; v < 8; ++v)
        C[(long)(r0 + v) * HF + col] = acc[v];
}

// ---------------------------------------------------------------- GCN conv
__global__ void k_deg(const int* __restrict__ dst, const float* __restrict__ m,
                      float* __restrict__ deg, int n) {
    long e = (long)blockIdx.x * blockDim.x + threadIdx.x;
    if (e >= EB) return;
    float mv = m[e];
    if (mv != 0.f) {
        int b = (int)(e / EE);
        atomicAdd(deg + b * n + dst[e], mv);
    }
}

__global__ void k_gcn_scatter(const int* __restrict__ src, const int* __restrict__ dst,
                              const float* __restrict__ m, const float* __restrict__ deg,
                              const float* __restrict__ xw, float* __restrict__ agg, int n) {
    long t = (long)blockIdx.x * blockDim.x + threadIdx.x;
    if (t >= (long)EB * HF) return;
    int e = (int)(t >> 6), h = (int)(t & 63);
    float mv = m[e];
    if (mv == 0.f) return;
    int b = e / EE;
    int s = b * n + src[e];
    int d = b * n + dst[e];
    float norm = rsqrtf(deg[s]) * rsqrtf(deg[d]) * mv;
    atomicAdd(agg + (long)d * HF + h, xw[(long)s * HF + h] * norm);
}

__global__ void k_gcn_finish(const float* __restrict__ agg, const float* __restrict__ xw,
                             const float* __restrict__ deg, const float* __restrict__ bias,
                             float* __restrict__ xc, long total) {
    long t = (long)blockIdx.x * blockDim.x + threadIdx.x;
    if (t >= total) return;
    long i = t >> 6;
    int h = (int)(t & 63);
    float v = agg[t] + xw[t] / deg[i] + bias[h];   // self-loop: xw * dis^2 = xw/deg
    xc[t] = v > 0.f ? v : 0.f;                     // relu
}

// ---------------------------------------------------------------- SAGPool
__global__ void k_neigh_scatter(const int* __restrict__ src, const int* __restrict__ dst,
                                const float* __restrict__ m, const float* __restrict__ xc,
                                float* __restrict__ ng, int n) {
    long t = (long)blockIdx.x * blockDim.x + threadIdx.x;
    if (t >= (long)EB * HF) return;
    int e = (int)(t >> 6), h = (int)(t & 63);
    float mv = m[e];
    if (mv == 0.f) return;
    int b = e / EE;
    atomicAdd(ng + (long)(b * n + dst[e]) * HF + h,
              xc[(long)(b * n + src[e]) * HF + h] * mv);
}

__global__ void k_score(const float* __restrict__ xc, const float* __restrict__ ng,
                        const float* __restrict__ Wr, const float* __restrict__ Wn,
                        const float* __restrict__ pb, float* __restrict__ score, int n) {
    int i = blockIdx.x * blockDim.x + threadIdx.x;
    if (i >= BB * n) return;
    const float* xr = xc + (long)i * HF;
    const float* nr = ng + (long)i * HF;
    float s = pb[0];
#pragma unroll 8
    for (int h = 0; h < HF; ++h) s += xr[h] * Wr[h] + nr[h] * Wn[h];
    score[i] = tanhf(s);
}

// per-graph top-k via full bitonic sort in LDS (n in {2048,1024,512}, pow2)
__global__ __launch_bounds__(1024) void k_topk(const float* __restrict__ score,
                                               int* __restrict__ perm, float* __restrict__ val,
                                               int* __restrict__ inv, int n, int k) {
    __shared__ float sv[2048];
    __shared__ int   si[2048];
    int b = blockIdx.x, tid = threadIdx.x;
    for (int i = tid; i < n; i += blockDim.x) { sv[i] = score[b * n + i]; si[i] = i; }
    __syncthreads();
    for (int size = 2; size <= n; size <<= 1) {
        for (int stride = size >> 1; stride > 0; stride >>= 1) {
            for (int t = tid; t < (n >> 1); t += blockDim.x) {
                int i = ((t / stride) * (stride << 1)) + (t % stride);
                int j = i + stride;
                float a = sv[i], c = sv[j];
                int ia = si[i], ic = si[j];
                bool a_first = (a > c) || (a == c && ia < ic); // descending, low idx on tie
                bool desc = ((i & size) == 0);
                if (desc ? !a_first : a_first) {
                    sv[i] = c; sv[j] = a; si[i] = ic; si[j] = ia;
                }
            }
            __syncthreads();
        }
    }
    for (int t = tid; t < n; t += blockDim.x) {
        int orig = si[t];
        inv[b * n + orig] = (t < k) ? t : -1;
        if (t < k) { perm[b * k + t] = orig; val[b * k + t] = sv[t]; }
    }
}

__global__ void k_gather(const float* __restrict__ xc, const int* __restrict__ perm,
                         const float* __restrict__ val, float* __restrict__ xn,
                         int n, int k) {
    long t = (long)blockIdx.x * blockDim.x + threadIdx.x;
    if (t >= (long)BB * k * HF) return;
    int h = (int)(t & 63);
    int slot = (int)(t >> 6);        // b*k + j
    int b = slot / k;
    int orig = perm[slot];
    xn[t] = xc[(long)(b * n + orig) * HF + h] * val[slot];
}

__global__ void k_remap(int* __restrict__ src, int* __restrict__ dst,
                        float* __restrict__ m, const int* __restrict__ inv, int n) {
    long e = (long)blockIdx.x * blockDim.x + threadIdx.x;
    if (e >= EB) return;
    int b = (int)(e / EE);
    int ns = inv[b * n + src[e]];
    int nd = inv[b * n + dst[e]];
    bool valid = (ns >= 0) && (nd >= 0) && (m[e] > 0.f);
    src[e] = valid ? ns : 0;
    dst[e] = valid ? nd : 0;
    m[e]   = valid ? 1.f : 0.f;
}

// -------------------------------------------------------------- readout/MLP
__global__ void k_readout(const float* __restrict__ xn, float* __restrict__ hacc, int k) {
    int t = blockIdx.x * blockDim.x + threadIdx.x;
    if (t >= BB * HF) return;
    int b = t >> 6, h = t & 63;
    const float* p = xn + (long)b * k * HF + h;
    float mx = -INFINITY, sm = 0.f;
    for (int j = 0; j < k; ++j) { float v = p[(long)j * HF]; mx = fmaxf(mx, v); sm += v; }
    hacc[b * 128 + h]      += mx;
    hacc[b * 128 + 64 + h] += sm;
}

__global__ __launch_bounds__(256) void k_mlp(const float* __restrict__ hacc,
        const float* __restrict__ hls,
        const float* __restrict__ W0, const float* __restrict__ b0,
        const float* __restrict__ W1, const float* __restrict__ b1,
        const float* __restrict__ W2, const float* __restrict__ b2,
        float* __restrict__ out) {
    __shared__ float hin[BB * 144];
    __shared__ float h1[BB * 64];
    __shared__ float h2[BB * 64];
    int tid = threadIdx.x;
    for (int t = tid; t < BB * 144; t += blockDim.x) {
        int r = t / 144, c = t % 144;
        hin[t] = (c < 128) ? hacc[r * 128 + c] : hls[r * 16 + (c - 128)];
    }
    __syncthreads();
    for (int t = tid; t < BB * 64; t += blockDim.x) {
        int r = t >> 6, c = t & 63;
        float s = b0[c];
        for (int kk = 0; kk < 144; ++kk) s += hin[r * 144 + kk] * W0[kk * 64 + c];
        h1[t] = s > 0.f ? s : 0.f;
    }
    __syncthreads();
    for (int t = tid; t < BB * 64; t += blockDim.x) {
        int r = t >> 6, c = t & 63;
        float s = b1[c];
        for (int kk = 0; kk < 64; ++kk) s += h1[r * 64 + kk] * W1[kk * 64 + c];
        h2[t] = s > 0.f ? s : 0.f;
    }
    __syncthreads();
    for (int t = tid; t < BB; t += blockDim.x) {
        float s = b2[0];
        for (int kk = 0; kk < 64; ++kk) s += h2[t * 64 + kk] * W2[kk];
        out[t] = s;
    }
}

// ------------------------------------------------------------------ driver
extern "C" void kernel_launch(void* const* d_in, const int* in_sizes, int n_in,
                              void* d_out, int out_size, void* d_ws, size_t ws_size,
                              hipStream_t stream) {
    (void)in_sizes; (void)n_in; (void)out_size; (void)ws_size;
    const float* x_in  = (const float*)d_in[0];
    const int*   ei    = (const int*)d_in[1];
    const float* hls   = (const float*)d_in[3];

    char* ws = (char*)d_ws;
    size_t off = 0;
    auto alloc = [&](size_t bytes) -> void* {
        void* p = ws + off;
        off = (off + bytes + 255) & ~(size_t)255;
        return p;
    };
    int*   srcb = (int*)  alloc((size_t)EB * 4);
    int*   dstb = (int*)  alloc((size_t)EB * 4);
    float* mb   = (float*)alloc((size_t)EB * 4);
    float* XW   = (float*)alloc((size_t)BB * NN * HF * 4);   // 16 MB
    float* AGG  = (float*)alloc((size_t)BB * NN * HF * 4);   // also reused as neigh
    float* XC   = (float*)alloc((size_t)BB * NN * HF * 4);
    float* XA   = (float*)alloc((size_t)BB * 1024 * HF * 4);
    float* XB2  = (float*)alloc((size_t)BB * 512 * HF * 4);
    float* deg  = (float*)alloc((size_t)BB * NN * 4);
    float* sc   = (float*)alloc((size_t)BB * NN * 4);
    int*   perm = (int*)  alloc((size_t)BB * 1024 * 4);
    float* val  = (float*)alloc((size_t)BB * 1024 * 4);
    int*   inv  = (int*)  alloc((size_t)BB * NN * 4);
    float* hacc = (float*)alloc((size_t)BB * 128 * 4);

    const int T = 256;
    k_edges<<<cdiv(EB, T), T, 0, stream>>>(ei, srcb, dstb, mb);
    k_fill<<<cdiv(BB * 128, T), T, 0, stream>>>(hacc, 0.f, BB * 128);

    const float* X = x_in;
    int n = NN, Kdim = CINW;
    float* xnext[3] = {XA, XB2, XA};

    for (int l = 0; l < 3; ++l) {
        const float* Wc = (const float*)d_in[4 + l];
        const float* bc = (const float*)d_in[7 + l];
        const float* Wr = (const float*)d_in[10 + l];
        const float* Wn = (const float*)d_in[13 + l];
        const float* pb = (const float*)d_in[16 + l];
        int M = BB * n;
        int k = n >> 1;
        long nodef = (long)M * HF;

        // GCNConv
        k_gemm_wmma<<<M / 32, 256, 0, stream>>>(X, Wc, XW, M, Kdim);
        k_fill<<<cdiv(M, T), T, 0, stream>>>(deg, 1.0f, M);
        k_deg<<<cdiv(EB, T), T, 0, stream>>>(dstb, mb, deg, n);
        k_fill<<<cdiv(nodef, T), T, 0, stream>>>(AGG, 0.f, nodef);
        k_gcn_scatter<<<cdiv((long)EB * HF, T), T, 0, stream>>>(srcb, dstb, mb, deg, XW, AGG, n);
        k_gcn_finish<<<cdiv(nodef, T), T, 0, stream>>>(AGG, XW, deg, bc, XC, nodef);

        // SAGPool
        k_fill<<<cdiv(nodef, T), T, 0, stream>>>(AGG, 0.f, nodef);   // AGG reused as neigh
        k_neigh_scatter<<<cdiv((long)EB * HF, T), T, 0, stream>>>(srcb, dstb, mb, XC, AGG, n);
        k_score<<<cdiv(M, T), T, 0, stream>>>(XC, AGG, Wr, Wn, pb, sc, n);
        k_topk<<<BB, 1024, 0, stream>>>(sc, perm, val, inv, n, k);
        k_gather<<<cdiv((long)BB * k * HF, T), T, 0, stream>>>(XC, perm, val, xnext[l], n, k);
        k_remap<<<cdiv(EB, T), T, 0, stream>>>(srcb, dstb, mb, inv, n);

        // readout: h += [max || sum]
        k_readout<<<cdiv(BB * HF, T), T, 0, stream>>>(xnext[l], hacc, k);

        X = xnext[l];
        n = k;
        Kdim = HF;
    }

    k_mlp<<<1, 256, 0, stream>>>(hacc, hls,
                                 (const float*)d_in[19], (const float*)d_in[22],
                                 (const float*)d_in[20], (const float*)d_in[23],
                                 (const float*)d_in[21], (const float*)d_in[24],
                                 (float*)d_out);
}